// BilateralFilter_27822798143865
// MI455X (gfx1250) — compile-verified
//
#include <hip/hip_runtime.h>
#include <stdint.h>

// Problem constants (from reference)
#define NB    8
#define HH    480
#define WW    640
#define KK    5
#define PAD   2

// Tiling: 64x16 outputs per 256-thread block (8 wave32s)
#define TILE_W 64
#define TILE_H 16
#define LW (TILE_W + 2*PAD)          // 68
#define LH (TILE_H + 2*PAD)          // 20
#define LDS_ELEMS (LW * LH)          // 1360

// exp(-d2/200) == exp2(d2 * (-log2(e)/200))
#define EXP2_SCALE (-0.00721347560222408455f)

typedef unsigned int u32;
typedef __attribute__((ext_vector_type(4))) u32 v4u;
typedef __attribute__((ext_vector_type(8))) int v8i;
typedef __attribute__((ext_vector_type(4))) int v4i;

#if __has_builtin(__builtin_amdgcn_tensor_load_to_lds) && __has_builtin(__builtin_amdgcn_s_wait_tensorcnt)
#define HAVE_TDM 1
#endif

// ---- per-lane async global->LDS dword copy (ASYNCcnt) ----
__device__ __forceinline__ void async_g2l_b32(const float* gptr, float* lptr) {
#if __has_builtin(__builtin_amdgcn_global_load_async_to_lds_b32)
    __builtin_amdgcn_global_load_async_to_lds_b32(
        (__attribute__((address_space(1))) int*)(gptr),
        (__attribute__((address_space(3))) int*)(lptr),
        0, 0);
#else
    unsigned lds_addr = (unsigned)(uintptr_t)lptr;
    asm volatile("global_load_async_to_lds_b32 %0, %1, off"
                 :: "v"(lds_addr), "v"(gptr) : "memory");
#endif
}

__device__ __forceinline__ void wait_async0() {
#if __has_builtin(__builtin_amdgcn_s_wait_asynccnt)
    __builtin_amdgcn_s_wait_asynccnt(0);
#else
    asm volatile("s_wait_asynccnt 0" ::: "memory");
#endif
}

__device__ __forceinline__ float gauss_w(float d2) {
#if __has_builtin(__builtin_amdgcn_exp2f)
    return __builtin_amdgcn_exp2f(d2 * EXP2_SCALE);   // one v_exp_f32
#else
    return __expf(d2 * -0.005f);
#endif
}

__global__ __launch_bounds__(256)
void BilateralFilter_27822798143865_kernel(const float* __restrict__ I,
                                           const float* __restrict__ g,
                                           float* __restrict__ out) {
    __shared__ float tile[LDS_ELEMS];
    __shared__ float gk[32];

    const int tx = threadIdx.x;            // 0..63
    const int ty = threadIdx.y;            // 0..3
    const int t  = ty * TILE_W + tx;       // 0..255 ; wave 0 == t<32
    const int x0 = blockIdx.x * TILE_W;
    const int y0 = blockIdx.y * TILE_H;
    const int n  = blockIdx.z;
    const float* In = I + (size_t)n * HH * WW;

    // Zero-init LDS: reference's zero padding + makes TDM OOB handling moot.
    for (int i = t; i < LDS_ELEMS; i += 256) tile[i] = 0.0f;
    if (t < 32) gk[t] = 0.0f;
    __syncthreads();

#ifdef HAVE_TDM
    const bool use_tdm = (blockIdx.x >= 1);   // left edge needs negative x start
#else
    const bool use_tdm = false;
#endif

    if (use_tdm) {
#ifdef HAVE_TDM
        if (t < 32) {  // wave 0 only: one DMA per block
            // 25 spatial-Gaussian coefficients (g tiled over H,W -> read (j,0,0))
            if (t < KK * KK) async_g2l_b32(g + (size_t)t * HH * WW, &gk[t]);

            const int gxs = x0 - PAD;                         // >= 0 since bx>=1
            const int gys = (y0 - PAD < 0) ? 0 : (y0 - PAD);  // clip top
            const int lyo = gys - (y0 - PAD);                 // 0 or 2 rows
            const u32 lds_base = (u32)(uintptr_t)&tile[0] + (u32)(lyo * LW * 4);
            const unsigned long long ga =
                (unsigned long long)(uintptr_t)(In + (size_t)gys * WW + gxs);
            const u32 td0 = (u32)(WW - gxs);   // cols left in tensor: right-edge OOB -> 0
            const u32 td1 = (u32)(HH - gys);   // rows left in tensor: bottom OOB -> 0
            const u32 tl0 = (u32)LW;           // 68 cols -> LDS rows contiguous at LW
            const u32 tl1 = (u32)(LH - lyo);   // 18 or 20 rows

            // D# group 0: count=1 | lds_addr | global_addr[56:0] | type=2
            v4u d0;
            d0[0] = 1u;
            d0[1] = lds_base;
            d0[2] = (u32)ga;
            d0[3] = (u32)((ga >> 32) & 0x01FFFFFFu) | (2u << 30);
            // D# group 1: data_size=4B; dims/strides in elements
            v8i d1;
            d1[0] = (int)(2u << 16);                                   // data_size=4B
            d1[1] = (int)((td0 & 0xFFFFu) << 16);                      // tensor_dim0[15:0]
            d1[2] = (int)(((td0 >> 16) & 0xFFFFu) | ((td1 & 0xFFFFu) << 16));
            d1[3] = (int)(((td1 >> 16) & 0xFFFFu) | (tl0 << 16));      // tile_dim0
            d1[4] = (int)tl1;                                          // tile_dim1 (tile_dim2=0)
            d1[5] = WW;                                                // tensor_dim0_stride
            d1[6] = 0;
            d1[7] = 0;
            v4i d2 = {0, 0, 0, 0};
            v4i d3 = {0, 0, 0, 0};
            v8i d4 = {0, 0, 0, 0, 0, 0, 0, 0};
            __builtin_amdgcn_tensor_load_to_lds(d0, d1, d2, d3, d4, 0);
            __builtin_amdgcn_s_wait_tensorcnt(0);
            wait_async0();
        }
#endif
    } else {
        // Left-edge blocks (or no TDM builtin): per-lane async staging.
        if (t < KK * KK) async_g2l_b32(g + (size_t)t * HH * WW, &gk[t]);
        for (int i = t; i < LDS_ELEMS; i += 256) {
            const int ly = i / LW;
            const int lx = i - ly * LW;
            const int gy = y0 - PAD + ly;
            const int gx = x0 - PAD + lx;
            if (gy >= 0 && gy < HH && gx >= 0 && gx < WW)
                async_g2l_b32(In + (size_t)gy * WW + gx, &tile[i]);
        }
        wait_async0();
    }
    __syncthreads();   // all staging visible block-wide

    // Each thread computes 4 output rows (16 rows / blockDim.y==4).
#pragma unroll
    for (int r = 0; r < 4; ++r) {
        const int yy = ty * 4 + r;                         // 0..15 in tile
        const float c = tile[(yy + PAD) * LW + (tx + PAD)];
        float wsum = 0.0f, acc = 0.0f;
#pragma unroll
        for (int dy = 0; dy < KK; ++dy) {
#pragma unroll
            for (int dx = 0; dx < KK; ++dx) {
                const float s = tile[(yy + dy) * LW + (tx + dx)];
                const float d = s - c;
                const float w = gauss_w(d * d) * gk[dy * KK + dx];
                wsum += w;
                acc  += w * s;
            }
        }
        out[((size_t)n * HH + (y0 + yy)) * WW + (x0 + tx)] = acc / wsum;
    }
}

extern "C" void kernel_launch(void* const* d_in, const int* in_sizes, int n_in,
                              void* d_out, int out_size, void* d_ws, size_t ws_size,
                              hipStream_t stream) {
    (void)in_sizes; (void)n_in; (void)d_ws; (void)ws_size; (void)out_size;
    const float* I = (const float*)d_in[0];   // (8,1,480,640) f32
    const float* g = (const float*)d_in[1];   // (1,25,480,640) f32 (tiled)
    float* out = (float*)d_out;               // (8,480,640) f32

    dim3 block(TILE_W, 4, 1);                         // 256 threads = 8 wave32
    dim3 grid(WW / TILE_W, HH / TILE_H, NB);          // 10 x 30 x 8
    BilateralFilter_27822798143865_kernel<<<grid, block, 0, stream>>>(I, g, out);
}